// PerceptualLinearPredictiveCoefficientsAnalysis_80058190397796
// MI455X (gfx1250) — compile-verified
//
#include <hip/hip_runtime.h>
#include <math.h>

typedef float v2f __attribute__((ext_vector_type(2)));
typedef float v8f __attribute__((ext_vector_type(8)));

#define NBINS   257
#define KDIM    256      // bins 0..255 ; H row 0 is zero (reference uses bins 1..255)
#define NCH     40
#define ORDER   16
#define FBSTR   49       // fb staging row stride (conflict-friendly)
#define MY_PI   3.14159265358979323846
// fragment-layout H: [kk(64)][tile(3)][lane(32)][elem(2)] floats = 12288 floats (48 KB)
#define HB_FLOATS (64 * 3 * 32 * 2)

__global__ __launch_bounds__(128)
void plp_kernel(const float* __restrict__ x, float* __restrict__ out, int nFrames)
{
    // ---- LDS: H in WMMA-B-fragment layout (reused later as fb staging) ----
    __shared__ __align__(16) float sH[HB_FLOATS];  // 48 KB
    __shared__ float sEql[NCH];
    __shared__ float sCos[(ORDER + 1) * NCH];      // cos(pi*(j+1)*k/41)

    const int tid  = threadIdx.x;
    const int wave = tid >> 5;
    const int lane = tid & 31;
    const int laneLo = lane & 15;
    const bool hi = lane >= 16;

    // ================= Phase 0: build constants in LDS =================
    for (int i = tid; i < HB_FLOATS; i += 128) sH[i] = 0.0f;

    const double mel_max = 1127.0 * log1p(8000.0 / 700.0); // F_MAX mel, mel_min = 0
    const double step = mel_max / 41.0;                    // uniform triangle spacing

    for (int b = 1 + tid; b < 256; b += 128) {
        double hz  = (16000.0 * (double)b) / 512.0;
        double mel = 1127.0 * log1p(hz / 700.0);
        double q   = mel / step;
        int ch = (int)ceil(q) - 1;            // searchsorted(cf_mel, mel, 'left')
        if (ch < 0) ch = 0;
        if (ch > 40) ch = 40;
        float w = (float)((step * (double)(ch + 1) - mel) / step);
        // fragment-layout scatter: (bin b, chan n) -> [kk][t][laneIdx][elem]
        const int kk = b >> 2, r = b & 3;
        const int laneHiOff = (r & 2) ? 16 : 0;
        const int elem = r & 1;
        if (ch >= 1) {
            int n = ch - 1;
            sH[(((kk * 3 + (n >> 4)) * 32) + laneHiOff + (n & 15)) * 2 + elem] = w;
        }
        if (ch <= 39) {
            int n = ch;
            sH[(((kk * 3 + (n >> 4)) * 32) + laneHiOff + (n & 15)) * 2 + elem] = 1.0f - w;
        }
    }
    if (tid < NCH) {
        double cf_hz = 700.0 * expm1(step * (double)(tid + 1) / 1127.0);
        double f2 = cf_hz * cf_hz;
        double e = (f2 / (f2 + 160000.0));
        e = e * e * (f2 + 1440000.0) / (f2 + 9610000.0);
        sEql[tid] = (float)e;
    }
    for (int i = tid; i < (ORDER + 1) * NCH; i += 128) {
        int k = i / NCH, j = i % NCH;
        sCos[i] = (float)cos(MY_PI * (double)((j + 1) * k) / 41.0);
    }
    __syncthreads();

    // ================= Phase 1: filterbank matmul via f32 WMMA =================
    // wave owns 32 frames: 2 M-tiles x 3 N-tiles, K = 256 in 64 steps of 4
    long frameBase = (long)blockIdx.x * 128 + (long)wave * 32;
    long f0 = frameBase + laneLo;        // A-row frame for M-tile 0
    long f1 = frameBase + 16 + laneLo;   // A-row frame for M-tile 1
    long fmax = (long)nFrames - 1;
    if (f0 > fmax) f0 = fmax;
    if (f1 > fmax) f1 = fmax;
    const float* __restrict__ row0 = x + (size_t)f0 * NBINS;
    const float* __restrict__ row1 = x + (size_t)f1 * NBINS;
    const int kShift = hi ? 2 : 0;

    // per-lane base into fragment-layout H: each k-step is 96 v2f (3 tiles x 32 lanes)
    const v2f* __restrict__ hb = ((const v2f*)sH) + lane;

    v8f acc[2][3];
    #pragma unroll
    for (int m = 0; m < 2; ++m)
        #pragma unroll
        for (int t = 0; t < 3; ++t)
            acc[m][t] = (v8f){0.f,0.f,0.f,0.f,0.f,0.f,0.f,0.f};

    for (int kk = 0; kk < 64; ++kk) {
        const int kb = kk * 4 + kShift;
        if ((kk & 15) == 0 && kk < 48) {  // uniform-branch streaming prefetch
            __builtin_prefetch(row0 + kb + 64, 0, 0);
            __builtin_prefetch(row1 + kb + 64, 0, 0);
        }
        v2f a0, a1;
        a0.x = row0[kb];     a0.y = row0[kb + 1];
        a1.x = row1[kb];     a1.y = row1[kb + 1];
        // single aligned 8B LDS load per B fragment (conflict-free: 64 banks)
        v2f b0 = hb[kk * 96];
        v2f b1 = hb[kk * 96 + 32];
        v2f b2 = hb[kk * 96 + 64];

        acc[0][0] = __builtin_amdgcn_wmma_f32_16x16x4_f32(false, a0, false, b0, (short)0, acc[0][0], false, false);
        acc[0][1] = __builtin_amdgcn_wmma_f32_16x16x4_f32(false, a0, false, b1, (short)0, acc[0][1], false, false);
        acc[0][2] = __builtin_amdgcn_wmma_f32_16x16x4_f32(false, a0, false, b2, (short)0, acc[0][2], false, false);
        acc[1][0] = __builtin_amdgcn_wmma_f32_16x16x4_f32(false, a1, false, b0, (short)0, acc[1][0], false, false);
        acc[1][1] = __builtin_amdgcn_wmma_f32_16x16x4_f32(false, a1, false, b1, (short)0, acc[1][1], false, false);
        acc[1][2] = __builtin_amdgcn_wmma_f32_16x16x4_f32(false, a1, false, b2, (short)0, acc[1][2], false, false);
    }

    // all waves are done reading sH -> reuse the region as fb staging
    __syncthreads();
    float* fbS = sH + wave * (32 * FBSTR);

    // C/D layout: VGPR j -> M = j (+8 for lanes 16..31), N = lane&15 (+16*t)
    #pragma unroll
    for (int m = 0; m < 2; ++m)
        #pragma unroll
        for (int t = 0; t < 3; ++t)
            #pragma unroll
            for (int j = 0; j < 8; ++j) {
                int rowi = m * 16 + j + (hi ? 8 : 0);
                int col  = t * 16 + laneLo;
                fbS[rowi * FBSTR + col] = acc[m][t][j];
            }

    // ================= Phase 2: per-frame scalar pipeline (1 lane = 1 frame) ===
    long frame = frameBase + lane;
    const float* myfb = fbS + lane * FBSTR;

    float y[NCH];
    #pragma unroll
    for (int j = 0; j < NCH; ++j) {
        float v = fmaxf(myfb[j], 1e-5f) * sEql[j];
        y[j] = powf(v, 0.33f);
    }

    // r[k] = (y0 + (-1)^k y39 + 2*sum_j y[j]*cos(pi*(j+1)*k/41)) / 82
    float r[ORDER + 1];
    #pragma unroll
    for (int k = 0; k <= ORDER; ++k) {
        float s = 0.0f;
        #pragma unroll
        for (int j = 0; j < NCH; ++j) s += y[j] * sCos[k * NCH + j];
        float ends = y[0] + ((k & 1) ? -y[NCH - 1] : y[NCH - 1]);
        r[k] = (ends + 2.0f * s) * (1.0f / 82.0f);
    }

    // Levinson-Durbin, order 16 (fully unrolled, registers)
    float a[ORDER];
    float E = r[0];
    {
        float k1 = -r[1] / E;
        a[0] = k1;
        E *= (1.0f - k1 * k1);
    }
    #pragma unroll
    for (int i = 2; i <= ORDER; ++i) {
        float av = r[i];
        #pragma unroll
        for (int j = 0; j <= i - 2; ++j) av += a[j] * r[i - 1 - j];
        float k = -av / E;
        float at[ORDER];
        #pragma unroll
        for (int j = 0; j <= i - 2; ++j) at[j] = a[j] + k * a[i - 2 - j];
        #pragma unroll
        for (int j = 0; j <= i - 2; ++j) a[j] = at[j];
        a[i - 1] = k;
        E *= (1.0f - k * k);
    }

    // LPC -> cepstrum (c[0]=log K is discarded by the reference; never needed)
    float c[ORDER + 1];
    c[1] = -a[0];
    #pragma unroll
    for (int m = 2; m <= ORDER; ++m) {
        float av = a[m - 1];
        #pragma unroll
        for (int k2 = 1; k2 < m; ++k2)
            av += ((float)k2 / (float)m) * c[k2] * a[m - k2 - 1];
        c[m] = -av;
    }

    if (frame < (long)nFrames) {
        float* op = out + (size_t)frame * ORDER;
        #pragma unroll
        for (int m = 1; m <= ORDER; ++m) {
            float lift = 1.0f + 11.0f * sinf((float)(MY_PI / 22.0) * (float)m);
            op[m - 1] = c[m] * lift;
        }
    }
}

extern "C" void kernel_launch(void* const* d_in, const int* in_sizes, int n_in,
                              void* d_out, int out_size, void* d_ws, size_t ws_size,
                              hipStream_t stream)
{
    (void)n_in; (void)d_ws; (void)ws_size; (void)out_size;
    const float* x = (const float*)d_in[0];
    float* out = (float*)d_out;
    int nFrames = in_sizes[0] / NBINS;           // 32*4096 = 131072
    int nBlocks = (nFrames + 127) / 128;         // 128 frames per block (4 waves x 32)
    plp_kernel<<<nBlocks, 128, 0, stream>>>(x, out, nFrames);
}